// UResidualMambaChannel_68564857913705
// MI455X (gfx1250) — compile-verified
//
#include <hip/hip_runtime.h>

// ---------------------------------------------------------------------------
// UResidualMambaChannel forward for MI455X (gfx1250, wave32).
// Layout convention: every activation tensor is (B, C, L) with L contiguous,
// so all reference transposes are layout no-ops. Channel-mixing matmuls use
// V_WMMA_F32_16X16X4_F32 (f32 in, f32 acc -> matches f32 reference numerics;
// problem is L2-resident / latency bound so low-precision WMMA buys nothing).
// The sequential SSM scan stages its strided B/C/x operands through LDS with
// the Tensor Data Mover (tensor_load_to_lds, double buffered, TENSORcnt).
// ---------------------------------------------------------------------------

#define LSEQ   2048
#define NBATCH 4
#define DSTATE 128

typedef __attribute__((ext_vector_type(2))) float v2f;
typedef __attribute__((ext_vector_type(8))) float v8f;

#ifdef __HIP_DEVICE_COMPILE__
#  if __has_builtin(__builtin_amdgcn_tensor_load_to_lds)
#    define USE_TDM 1
#  else
#    define USE_TDM 0
#  endif
#else
#  define USE_TDM 0
#endif

// LDS staging geometry for the scan: chunks of 32 timesteps, rows = 16 x-rows
// followed by 128 B-rows + 128 C-rows, LDS row stride 35 dwords (32 data + 3
// pad from the TDM pad feature -> avoids 32-way LDS bank conflicts).
#define SCAN_T        32
#define SCAN_RSTRIDE  35
#define SCAN_XROWS    16
#define SCAN_BCROWS   256
#define SCAN_BUFDW    ((SCAN_XROWS + SCAN_BCROWS) * SCAN_RSTRIDE)   // 9520 dwords
#define SCAN_SMEM     (2 * SCAN_BUFDW * 4)                          // 76160 bytes

__device__ __forceinline__ float silu_f(float x) {
  return x * (1.0f / (1.0f + __expf(-x)));
}

#if USE_TDM
typedef unsigned int v4u __attribute__((ext_vector_type(4)));
typedef int          v4i __attribute__((ext_vector_type(4)));
typedef int          v8i __attribute__((ext_vector_type(8)));

// 2-D tile load: nrows rows of SCAN_T f32 elements from a row-major tensor with
// row stride LSEQ, into LDS with 3-dword padding after every 32 dwords.
__device__ __forceinline__ void tdm_load_2d(unsigned int lds_byte,
                                            const float* gaddr, int nrows) {
  unsigned long long ga = (unsigned long long)(size_t)gaddr;
  v4u g0;
  g0[0] = 1u;                                   // count=1, user descriptor
  g0[1] = lds_byte;                             // lds_addr
  g0[2] = (unsigned int)(ga & 0xffffffffu);     // global_addr[31:0]
  g0[3] = (unsigned int)((ga >> 32) & 0x1ffffffu) | 0x80000000u; // ga[56:32] | type=2
  v8i g1;
  g1[0] = 0x05120000;        // data_size=4B, pad_enable, pad_interval=32dw, pad_amount=3dw
  g1[1] = 0x08000000;        // tensor_dim0 = 2048 (bits 79:48, low half)
  g1[2] = nrows << 16;       // tensor_dim1 (bits 111:80, low half)
  g1[3] = SCAN_T << 16;      // tile_dim0 (bits 127:112)
  g1[4] = nrows;             // tile_dim1 (bits 143:128); tile_dim2 = 0
  g1[5] = LSEQ;              // tensor_dim0_stride (bits 207:160, low 32)
  g1[6] = 0;
  g1[7] = 0;
  v4i z4 = {0, 0, 0, 0};                  // groups 2/3 unused (<= 2D tile)
  v8i z8 = {0, 0, 0, 0, 0, 0, 0, 0};      // extra group (6-arg toolchain variant)
  __builtin_amdgcn_tensor_load_to_lds(g0, g1, z4, z4, z8, 0);
}
#endif

// ---------------------------------------------------------------------------
// Channel GEMM: out[b, outC0+o, l] = act( sum_k W[o,k] * in[b, inC0+k, l] + bias[o] ) + add[...]
// One wave computes a 16(M=o) x 64(N=l) tile; K stepped by 4 with WMMA f32 16x16x4.
// A fragment: lanes 0-15 -> {K, K+1}, lanes 16-31 -> {K+2, K+3} (ISA 7.12.2).
// Out-of-range output rows: clamp the W row (stays in-bounds) and scale by a
// 0/1 mask -> unconditional loads, no EXEC save/restore in the hot loop.
// K must be a multiple of 4 (true for every matmul in this model).
// ---------------------------------------------------------------------------
__global__ void __launch_bounds__(256) k_gemm(
    const float* __restrict__ W, const float* __restrict__ bias,
    const float* __restrict__ in, long inSB, int inC0,
    float* __restrict__ out, long outSB, int outC0,
    const float* __restrict__ add, long adSB, int adC0,
    int O, int K, int act, int nOT, int nWaves)
{
  int wid = (int)((blockIdx.x * blockDim.x + threadIdx.x) >> 5);
  if (wid >= nWaves) return;                 // uniform per wave: EXEC stays all-ones for WMMA
  int lane = (int)(threadIdx.x & 31);

  const int nLT = LSEQ / 64;
  int lt   = wid % nLT;
  int rest = wid / nLT;
  int ot   = rest % nOT;
  int b    = rest / nOT;

  int mrow = lane & 15;
  int hi   = lane >> 4;                      // 0 or 1
  int o0   = ot * 16;
  long lbase = (long)lt * 64 + mrow;

  const float* inB = in + (long)b * inSB + (long)inC0 * LSEQ;
  int   arow  = o0 + mrow;
  float amask = (arow < O) ? 1.0f : 0.0f;
  int   arc   = (arow < O) ? arow : (O - 1);
  const float* wrow = W + (long)arc * K;

  v8f acc[4];
  #pragma unroll
  for (int t = 0; t < 4; ++t)
    #pragma unroll
    for (int v = 0; v < 8; ++v) acc[t][v] = 0.0f;

  for (int k = 0; k < K; k += 4) {
    v2f a;
    a.x = wrow[k + 2 * hi]     * amask;
    a.y = wrow[k + 2 * hi + 1] * amask;
    const float* bp = inB + (long)(k + 2 * hi) * LSEQ + lbase;
    __builtin_prefetch(bp + 4 * (long)LSEQ, 0, 1);   // next K-step rows -> global_prefetch_b8
    __builtin_prefetch(bp + 5 * (long)LSEQ, 0, 1);
    #pragma unroll
    for (int t = 0; t < 4; ++t) {
      v2f bb;
      bb.x = bp[t * 16];
      bb.y = bp[t * 16 + LSEQ];
      acc[t] = __builtin_amdgcn_wmma_f32_16x16x4_f32(
          false, a, false, bb, (short)0, acc[t], false, false);
    }
  }

  // Epilogue. C/D layout: VGPR v -> row (v + 8*hi), col = lane&15.
  #pragma unroll
  for (int t = 0; t < 4; ++t) {
    long col = (long)lt * 64 + t * 16 + mrow;
    #pragma unroll
    for (int v = 0; v < 8; ++v) {
      int orow = o0 + v + 8 * hi;
      if (orow < O) {
        float val = acc[t][v];
        if (bias) val += bias[orow];
        if (act)  val  = silu_f(val);
        if (add)  val += add[(long)b * adSB + (long)(adC0 + orow) * LSEQ + col];
        out[(long)b * outSB + (long)(outC0 + orow) * LSEQ + col] = val;
      }
    }
  }
}

// ---------------------------------------------------------------------------
// RMSNorm over channels for (B,C,L) layout, with optional SiLU gate, gamma and
// residual addend. One thread per (b,l); two passes over C (L2 resident).
// ---------------------------------------------------------------------------
__global__ void __launch_bounds__(256) k_rmsnorm(
    const float* __restrict__ in, long inSB, int inC0,
    const float* __restrict__ gate, long gSB, int gC0,
    const float* __restrict__ gamma,
    const float* __restrict__ add, long adSB, int adC0,
    float* __restrict__ out, long outSB, int outC0,
    int C, int total)
{
  int t = blockIdx.x * blockDim.x + threadIdx.x;
  if (t >= total) return;
  int b = t >> 11;              // L = 2048
  int l = t & (LSEQ - 1);

  const float* ip = in + (long)b * inSB + (long)inC0 * LSEQ + l;
  const float* gp = gate ? gate + (long)b * gSB + (long)gC0 * LSEQ + l : nullptr;
  float ss = 0.0f;
  for (int c = 0; c < C; ++c) {
    float v = ip[(long)c * LSEQ];
    if (gp) v *= silu_f(gp[(long)c * LSEQ]);
    ss += v * v;
  }
  float r = rsqrtf(ss / (float)C + 1e-5f);

  const float* ap = add ? add + (long)b * adSB + (long)adC0 * LSEQ + l : nullptr;
  float* op = out + (long)b * outSB + (long)outC0 * LSEQ + l;
  for (int c = 0; c < C; ++c) {
    float v = ip[(long)c * LSEQ];
    if (gp) v *= silu_f(gp[(long)c * LSEQ]);
    float o = v * r;
    if (gamma) o *= gamma[c];
    if (ap)    o += ap[(long)c * LSEQ];
    op[(long)c * LSEQ] = o;
  }
}

// dt[b,h,l] = softplus(zx[b, c0+h, l] + dt_bias[h])
__global__ void __launch_bounds__(256) k_dtsp(
    const float* __restrict__ zx, long SB, int c0,
    const float* __restrict__ dt_bias,
    float* __restrict__ dt, int nh, int total)
{
  int t = blockIdx.x * blockDim.x + threadIdx.x;
  if (t >= total) return;
  int l = t & (LSEQ - 1);
  int rest = t >> 11;
  int h = rest % nh;
  int b = rest / nh;
  float x = zx[(long)b * SB + (long)(c0 + h) * LSEQ + l] + dt_bias[h];
  float sp = (x > 20.0f) ? x : log1pf(__expf(x));
  dt[((long)b * nh + h) * LSEQ + l] = sp;
}

// Depthwise causal conv (D_CONV=4) + bias + SiLU over L, per channel.
__global__ void __launch_bounds__(256) k_dwconv(
    const float* __restrict__ in, long inSB, int inC0,
    const float* __restrict__ cw, const float* __restrict__ cb,
    float* __restrict__ out, long outSB,
    int C, int total)
{
  int t = blockIdx.x * blockDim.x + threadIdx.x;
  if (t >= total) return;
  int l = t & (LSEQ - 1);
  int rest = t >> 11;
  int c = rest % C;
  int b = rest / C;
  const float* ip = in + (long)b * inSB + (long)(inC0 + c) * LSEQ;
  float acc = cb[c];
  #pragma unroll
  for (int k = 0; k < 4; ++k) {
    int ls = l - 3 + k;
    if (ls >= 0) acc += cw[c * 4 + k] * ip[ls];
  }
  out[(long)b * outSB + (long)c * LSEQ + l] = silu_f(acc);
}

// ---------------------------------------------------------------------------
// Sequential SSM scan. One wave32 per (b, head). headdim=16, d_state=128.
// Lane owns d_state slice n in [lane*4, lane*4+4) for all 16 headdim rows ->
// 64 state VGPRs. Per step: h = h*dA + (dt*x_p)*B_n ; y_p = sum_n h*C_n via
// 5-stage shfl_xor butterfly; lane 0 stores y_p + D*x_p.
// TDM path: each 32-step chunk of the x(16 rows) and B|C(256 rows) operand
// tiles is DMA'd to LDS (tensor_load_to_lds, padded rows), double buffered
// and synchronized with s_wait_tensorcnt.
// ---------------------------------------------------------------------------
__global__ void __launch_bounds__(32) k_scan(
    const float* __restrict__ dt,
    const float* __restrict__ xbc, long xSB,
    const float* __restrict__ A_log, const float* __restrict__ Dp,
    float* __restrict__ y, long ySB, int nheads)
{
  int wid  = blockIdx.x;
  int lane = threadIdx.x;
  int b = wid / nheads;
  int h = wid % nheads;
  int d_inner = nheads * 16;

  float a  = -__expf(A_log[h]);
  float Dh = Dp[h];

  const float* xp  = xbc + (long)b * xSB + (long)(h * 16) * LSEQ;   // x rows (16)
  const float* bc0 = xbc + (long)b * xSB + (long)d_inner * LSEQ;    // B rows then C rows (256)
  const float* dtp = dt + ((long)b * nheads + h) * LSEQ;
  float* yp = y + (long)b * ySB + (long)(h * 16) * LSEQ;
  int lane4 = lane * 4;

  float hreg[64];
  #pragma unroll
  for (int i = 0; i < 64; ++i) hreg[i] = 0.0f;

#if USE_TDM
  extern __shared__ float sm[];
  unsigned int smbase = (unsigned int)(unsigned long long)(size_t)sm; // LDS byte address
  const int NCHUNK = LSEQ / SCAN_T;

  // prologue: fetch chunk 0 into buffer 0
  tdm_load_2d(smbase, xp, SCAN_XROWS);
  tdm_load_2d(smbase + SCAN_XROWS * SCAN_RSTRIDE * 4, bc0, SCAN_BCROWS);

  for (int ci = 0; ci < NCHUNK; ++ci) {
    if (ci + 1 < NCHUNK) {
      // all LDS reads of the buffer we are about to overwrite have completed
      asm volatile("s_wait_dscnt 0x0" ::: "memory");
      unsigned int base = smbase + (unsigned int)(((ci + 1) & 1) * SCAN_BUFDW * 4);
      int l0 = (ci + 1) * SCAN_T;
      tdm_load_2d(base, xp + l0, SCAN_XROWS);
      tdm_load_2d(base + SCAN_XROWS * SCAN_RSTRIDE * 4, bc0 + l0, SCAN_BCROWS);
      __builtin_amdgcn_s_wait_tensorcnt(2);   // chunk ci's two loads done
    } else {
      __builtin_amdgcn_s_wait_tensorcnt(0);
    }
    asm volatile("" ::: "memory");
    const float* xl  = sm + (ci & 1) * SCAN_BUFDW;
    const float* bcl = xl + SCAN_XROWS * SCAN_RSTRIDE;

    for (int t = 0; t < SCAN_T; ++t) {
      int l = ci * SCAN_T + t;
      float dtt = dtp[l];
      float dA  = __expf(dtt * a);
      float Bl[4], Cl[4];
      #pragma unroll
      for (int j = 0; j < 4; ++j) {
        Bl[j] = bcl[(lane4 + j) * SCAN_RSTRIDE + t];
        Cl[j] = bcl[(DSTATE + lane4 + j) * SCAN_RSTRIDE + t];
      }
      float accv[16];
      #pragma unroll
      for (int p = 0; p < 16; ++p) {
        float coeff = dtt * xl[p * SCAN_RSTRIDE + t];
        float s = 0.0f;
        #pragma unroll
        for (int j = 0; j < 4; ++j) {
          float hv = hreg[p * 4 + j] * dA + coeff * Bl[j];
          hreg[p * 4 + j] = hv;
          s += hv * Cl[j];
        }
        accv[p] = s;
      }
      #pragma unroll
      for (int p = 0; p < 16; ++p) {
        float v = accv[p];
        v += __shfl_xor(v, 1, 32);
        v += __shfl_xor(v, 2, 32);
        v += __shfl_xor(v, 4, 32);
        v += __shfl_xor(v, 8, 32);
        v += __shfl_xor(v, 16, 32);
        accv[p] = v;
      }
      if (lane == 0) {
        #pragma unroll
        for (int p = 0; p < 16; ++p)
          yp[(long)p * LSEQ + l] = accv[p] + Dh * xl[p * SCAN_RSTRIDE + t];
      }
    }
  }
#else
  const float* Bp = bc0 + (long)lane4 * LSEQ;
  const float* Cp = Bp + (long)DSTATE * LSEQ;
  for (int l = 0; l < LSEQ; ++l) {
    float dtt = dtp[l];
    float dA  = __expf(dtt * a);
    float Bl[4], Cl[4];
    #pragma unroll
    for (int j = 0; j < 4; ++j) {
      Bl[j] = Bp[(long)j * LSEQ + l];
      Cl[j] = Cp[(long)j * LSEQ + l];
    }
    float accv[16];
    #pragma unroll
    for (int p = 0; p < 16; ++p) {
      float coeff = dtt * xp[(long)p * LSEQ + l];
      float s = 0.0f;
      #pragma unroll
      for (int j = 0; j < 4; ++j) {
        float hv = hreg[p * 4 + j] * dA + coeff * Bl[j];
        hreg[p * 4 + j] = hv;
        s += hv * Cl[j];
      }
      accv[p] = s;
    }
    #pragma unroll
    for (int p = 0; p < 16; ++p) {
      float v = accv[p];
      v += __shfl_xor(v, 1, 32);
      v += __shfl_xor(v, 2, 32);
      v += __shfl_xor(v, 4, 32);
      v += __shfl_xor(v, 8, 32);
      v += __shfl_xor(v, 16, 32);
      accv[p] = v;
    }
    if (lane == 0) {
      #pragma unroll
      for (int p = 0; p < 16; ++p)
        yp[(long)p * LSEQ + l] = accv[p] + Dh * xp[(long)p * LSEQ + l];
    }
  }
#endif
}

__global__ void __launch_bounds__(256) k_mul(const float* __restrict__ a,
                                             const float* __restrict__ b,
                                             float* __restrict__ o, long n)
{
  long i = (long)blockIdx.x * blockDim.x + threadIdx.x;
  if (i < n) o[i] = a[i] * b[i];
}

__global__ void __launch_bounds__(256) k_add(const float* __restrict__ a,
                                             const float* __restrict__ b,
                                             float* __restrict__ o, long n)
{
  long i = (long)blockIdx.x * blockDim.x + threadIdx.x;
  if (i < n) o[i] = a[i] + b[i];
}

// ---------------------------------------------------------------------------
// Host side: parameter pytree parsing + orchestration.
// ---------------------------------------------------------------------------
namespace {

struct LinP { const float *b, *w; };
struct MamP { const float *A_log, *D, *conv_b, *conv_w, *dt_bias, *in_w, *norm_w, *out_w; };
struct EncP { LinP fp, ip; MamP m; LinP op, wp; };
struct LvlP { LinP conv; EncP enc; };

struct Reader {
  void* const* d;
  int i;
  bool sorted;   // true: dict keys flattened in sorted order; false: insertion order
  const float* next() { return (const float*)d[i++]; }
};

LinP rdLin(Reader& r) {
  LinP p;
  if (r.sorted) { p.b = r.next(); p.w = r.next(); }
  else          { p.w = r.next(); p.b = r.next(); }
  return p;
}

MamP rdMam(Reader& r) {
  MamP m;
  if (r.sorted) {
    m.A_log = r.next(); m.D = r.next(); m.conv_b = r.next(); m.conv_w = r.next();
    m.dt_bias = r.next(); m.in_w = r.next(); m.norm_w = r.next(); m.out_w = r.next();
  } else {
    m.in_w = r.next(); m.conv_w = r.next(); m.conv_b = r.next(); m.dt_bias = r.next();
    m.A_log = r.next(); m.D = r.next(); m.norm_w = r.next(); m.out_w = r.next();
  }
  return m;
}

EncP rdEnc(Reader& r) {
  EncP e;
  if (r.sorted) {
    e.fp = rdLin(r); e.ip = rdLin(r); e.m = rdMam(r); e.op = rdLin(r); e.wp = rdLin(r);
  } else {
    e.ip = rdLin(r); e.wp = rdLin(r); e.fp = rdLin(r); e.op = rdLin(r); e.m = rdMam(r);
  }
  return e;
}

} // namespace

extern "C" void kernel_launch(void* const* d_in, const int* in_sizes, int n_in,
                              void* d_out, int out_size, void* d_ws, size_t ws_size,
                              hipStream_t stream)
{
  (void)out_size; (void)ws_size;
  const int X_ELEMS = NBATCH * 64 * LSEQ;   // 524288, unique among all leaves

  // Locate x (top-level dict may be insertion- or sorted-flattened).
  const float* x = nullptr;
  int pstart = 0;
  if (in_sizes[0] == X_ELEMS) { x = (const float*)d_in[0]; pstart = 1; }
  else                        { x = (const float*)d_in[n_in - 1]; pstart = 0; }

  // Detect dict ordering: first two param leaves are downs[0].conv {b(32), w(2048)}.
  Reader r{d_in, pstart, in_sizes[pstart] < in_sizes[pstart + 1]};

  LvlP downs[4]; EncP mid[2]; LvlP ups[4];
  for (int i = 0; i < 4; ++i) { downs[i].conv = rdLin(r); downs[i].enc = rdEnc(r); }
  for (int i = 0; i < 2; ++i) { mid[i] = rdEnc(r); }
  for (int i = 0; i < 4; ++i) { ups[i].conv = rdLin(r); ups[i].enc = rdEnc(r); }

  // Workspace arena (float units). Total ~2200 channel-planes = ~72 MB.
  float* base = (float*)d_ws;
  size_t off = 0;
  auto arena = [&](int ch) { float* p = base + off; off += (size_t)ch * NBATCH * LSEQ; return p; };
  float* r1  = arena(32);    // residual after down level 0
  float* r2  = arena(16);
  float* r3  = arena(8);
  float* tA  = arena(64);    // generic ping
  float* tB  = arena(64);    // generic pong
  float* tn  = arena(64);    // normed encoder input
  float* sbuf= arena(128);   // mamba input / s path
  float* wbuf= arena(128);   // gate path w
  float* zx  = arena(784);   // zxbcdt
  float* xbc = arena(512);   // post-conv xBC
  float* dtb = arena(16);    // dt (B, nheads, L)
  float* ybuf= arena(256);   // ip temp + scan output + gated-norm (in place)
  float* mbuf= arena(128);   // mamba output

  auto gemm = [&](const float* W, const float* bias,
                  const float* in, long inSB, int inC0,
                  float* out, long outSB, int outC0,
                  const float* add, long adSB, int adC0,
                  int O, int K, int act) {
    int nOT = (O + 15) / 16;
    int nWaves = NBATCH * nOT * (LSEQ / 64);
    int thr = nWaves * 32;
    k_gemm<<<(thr + 255) / 256, 256, 0, stream>>>(
        W, bias, in, inSB, inC0, out, outSB, outC0, add, adSB, adC0,
        O, K, act, nOT, nWaves);
  };

  auto rms = [&](const float* in, long inSB, int inC0,
                 const float* gate, long gSB, int gC0,
                 const float* gamma,
                 const float* add, long adSB, int adC0,
                 float* out, long outSB, int outC0, int C) {
    int total = NBATCH * LSEQ;
    k_rmsnorm<<<(total + 255) / 256, 256, 0, stream>>>(
        in, inSB, inC0, gate, gSB, gC0, gamma, add, adSB, adC0,
        out, outSB, outC0, C, total);
  };

  // Encoder at channel count c (all tensors (B,*,L); transposes are no-ops).
  auto encoder = [&](const EncP& e, int c, const float* tin, float* tout) {
    int dm = 2 * c, di = 4 * c, nh = c / 4;
    int cdim = di + 2 * DSTATE;
    int dip  = 2 * di + 2 * DSTATE + nh;
    long cS = (long)c * LSEQ, dmS = (long)dm * LSEQ, diS = (long)di * LSEQ;
    long dipS = (long)dip * LSEQ, cdS = (long)cdim * LSEQ;

    // tn = rmsnorm(tin)
    rms(tin, cS, 0, nullptr, 0, 0, nullptr, nullptr, 0, 0, tn, cS, 0, c);
    // s = fp(ip(tn))     (ip temp staged in ybuf, free until scan)
    gemm(e.ip.w, e.ip.b, tn, cS, 0, ybuf, dmS, 0, nullptr, 0, 0, dm, c, 0);
    gemm(e.fp.w, e.fp.b, ybuf, dmS, 0, sbuf, dmS, 0, nullptr, 0, 0, dm, dm, 0);
    // w = silu(wp(tn))
    gemm(e.wp.w, e.wp.b, tn, cS, 0, wbuf, dmS, 0, nullptr, 0, 0, dm, c, 1);

    // ---- mamba2(s), d_model = dm ----
    gemm(e.m.in_w, nullptr, sbuf, dmS, 0, zx, dipS, 0, nullptr, 0, 0, dip, dm, 0);
    { int total = NBATCH * nh * LSEQ;
      k_dtsp<<<(total + 255) / 256, 256, 0, stream>>>(zx, dipS, di + cdim,
                                                      e.m.dt_bias, dtb, nh, total); }
    { int total = NBATCH * cdim * LSEQ;
      k_dwconv<<<(total + 255) / 256, 256, 0, stream>>>(zx, dipS, di,
                                                        e.m.conv_w, e.m.conv_b,
                                                        xbc, cdS, cdim, total); }
    k_scan<<<NBATCH * nh, 32, SCAN_SMEM, stream>>>(dtb, xbc, cdS, e.m.A_log, e.m.D,
                                                   ybuf, diS, nh);
    // y = rmsnorm(y * silu(z)) * norm_w   (z = zx channels [0, di); in place)
    rms(ybuf, diS, 0, zx, dipS, 0, e.m.norm_w, nullptr, 0, 0, ybuf, diS, 0, di);
    // m = out_proj(y)
    gemm(e.m.out_w, nullptr, ybuf, diS, 0, mbuf, dmS, 0, nullptr, 0, 0, dm, di, 0);
    // ---- end mamba ----

    // s = rmsnorm(m) + s   (in place on s)
    rms(mbuf, dmS, 0, nullptr, 0, 0, nullptr, sbuf, dmS, 0, sbuf, dmS, 0, dm);
    // w = w * s
    { long n = (long)NBATCH * dm * LSEQ;
      k_mul<<<(int)((n + 255) / 256), 256, 0, stream>>>(wbuf, sbuf, wbuf, n); }
    // tout = op(w) + tin
    gemm(e.op.w, e.op.b, wbuf, dmS, 0, tout, cS, 0, tin, cS, 0, c, dm, 0);
  };

  auto addk = [&](const float* a, const float* b, float* o, long n) {
    k_add<<<(int)((n + 255) / 256), 256, 0, stream>>>(a, b, o, n);
  };

  const long BL = (long)NBATCH * LSEQ;

  // ---- down path (residual[i] is the input of level i) ----
  gemm(downs[0].conv.w, downs[0].conv.b, x, 64L * LSEQ, 0, tA, 32L * LSEQ, 0,
       nullptr, 0, 0, 32, 64, 0);
  encoder(downs[0].enc, 32, tA, r1);
  gemm(downs[1].conv.w, downs[1].conv.b, r1, 32L * LSEQ, 0, tA, 16L * LSEQ, 0,
       nullptr, 0, 0, 16, 32, 0);
  encoder(downs[1].enc, 16, tA, r2);
  gemm(downs[2].conv.w, downs[2].conv.b, r2, 16L * LSEQ, 0, tA, 8L * LSEQ, 0,
       nullptr, 0, 0, 8, 16, 0);
  encoder(downs[2].enc, 8, tA, r3);
  gemm(downs[3].conv.w, downs[3].conv.b, r3, 8L * LSEQ, 0, tA, 4L * LSEQ, 0,
       nullptr, 0, 0, 4, 8, 0);
  encoder(downs[3].enc, 4, tA, tB);

  // ---- mid ----
  encoder(mid[0], 4, tB, tA);
  encoder(mid[1], 4, tA, tB);

  // ---- up path ----
  gemm(ups[0].conv.w, ups[0].conv.b, tB, 4L * LSEQ, 0, tA, 8L * LSEQ, 0,
       nullptr, 0, 0, 8, 4, 0);
  encoder(ups[0].enc, 8, tA, tB);
  addk(tB, r3, tA, 8 * BL);

  gemm(ups[1].conv.w, ups[1].conv.b, tA, 8L * LSEQ, 0, tB, 16L * LSEQ, 0,
       nullptr, 0, 0, 16, 8, 0);
  encoder(ups[1].enc, 16, tB, tA);
  addk(tA, r2, tB, 16 * BL);

  gemm(ups[2].conv.w, ups[2].conv.b, tB, 16L * LSEQ, 0, tA, 32L * LSEQ, 0,
       nullptr, 0, 0, 32, 16, 0);
  encoder(ups[2].enc, 32, tA, tB);
  addk(tB, r1, tA, 32 * BL);

  gemm(ups[3].conv.w, ups[3].conv.b, tA, 32L * LSEQ, 0, tB, 64L * LSEQ, 0,
       nullptr, 0, 0, 64, 32, 0);
  encoder(ups[3].enc, 64, tB, tA);
  addk(tA, x, (float*)d_out, 64 * BL);
}